// ImprovedNoncommutativeKAOperator_29180007809783
// MI455X (gfx1250) — compile-verified
//
#include <hip/hip_runtime.h>
#include <stdint.h>

// ---------------------------------------------------------------------------
// Reference collapses to: out = 256MB of complex128 zeros with a REAL diagonal
//   d_n = Re(n^{-s}) [+ tiny prime corrections] + reg,
//   reg = max(1e-20, 1e-16 * sqrt(sum d_n^2)).
// (Prime off-diagonals cancel exactly under 0.5*(H + H^dagger).)
// Memory-bound: ~268 MB of stores -> ~11.5us floor at 23.3 TB/s.
// Bulk zero fill uses the CDNA5 Tensor Data Mover (tensor_store_from_lds):
// one 32KB DMA per wave from a zeroed LDS tile, tracked via TENSORcnt.
// Store uses NT temporal hint (cpol TH=1): 256MB stream > 192MB L2.
// ---------------------------------------------------------------------------

#define DIM 4096
#define CHUNK_DOUBLES 4096               // 32 KB per TDM store
#define CHUNK_BYTES   (CHUNK_DOUBLES * 8)
#define TOTAL_DOUBLES ((size_t)DIM * DIM * 2)   // complex128 -> 2 doubles/elem
#define N_CHUNKS      (TOTAL_DOUBLES / CHUNK_DOUBLES)  // 8192

typedef __attribute__((ext_vector_type(4))) unsigned int v4u;
typedef __attribute__((ext_vector_type(8))) int          v8i;
typedef __attribute__((ext_vector_type(4))) int          v4i;

__constant__ int c_primes[50] = {
    2,3,5,7,11,13,17,19,23,29,31,37,41,43,47,53,59,61,67,71,
    73,79,83,89,97,101,103,107,109,113,127,131,137,139,149,151,157,163,167,173,
    179,181,191,193,197,199,211,223,227,229};

// --- Kernel 1: diagonal values, Frobenius norm, regularizer (deterministic) --
__global__ __launch_bounds__(1024) void k_diag(const double* __restrict__ s_real,
                                               const double* __restrict__ s_imag,
                                               double* __restrict__ ws) {
    const double sr = s_real[0];
    const double si = s_imag[0];
    const double corr_strength = fmin(sqrt(sr * sr + si * si), 10.0);
    const double ZETA2 = 1.6449340668482264365;  // pi^2/6
    const double THETA = 1e-22;

    __shared__ double red[1024];
    double local = 0.0;
    // Each thread owns 4 indices in a fixed order -> deterministic partials.
    for (int k = 0; k < 4; ++k) {
        int idx = (int)threadIdx.x + k * 1024;   // 0..4095, n = idx+1
        double n  = (double)(idx + 1);
        double ln = log(n);
        double re;
        if (-sr * ln > -100.0) {
            re = exp(-sr * ln) * cos(si * ln);
        } else {
            re = 1e-100;
        }
        if (idx + 1 <= 229) {                    // prime diagonal correction
            for (int j = 0; j < 50; ++j) {
                if (c_primes[j] == idx + 1) {
                    re += THETA * ln * corr_strength * (ZETA2 / n);
                }
            }
        }
        ws[idx] = re;
        local += re * re;
    }
    red[threadIdx.x] = local;
    __syncthreads();
    // Fixed-order tree reduction -> bit-deterministic.
    for (int off = 512; off > 0; off >>= 1) {
        if ((int)threadIdx.x < off) red[threadIdx.x] += red[threadIdx.x + off];
        __syncthreads();
    }
    if (threadIdx.x == 0) {
        double norm = sqrt(red[0]);
        ws[DIM] = fmax(1e-20, norm * 1e-16);
    }
}

// --- Kernel 2: zero fill of the whole 256MB output via TDM ------------------
// 1024 blocks x 256 threads = 8192 waves; each wave DMAs one 32KB zero chunk
// from LDS to global with a single tensor_store_from_lds (NT temporal hint).
__global__ __launch_bounds__(256) void k_zero_tdm(double* __restrict__ out) {
    __shared__ double zbuf[CHUNK_DOUBLES];  // 32 KB of zeros
    for (int i = (int)threadIdx.x; i < CHUNK_DOUBLES; i += (int)blockDim.x)
        zbuf[i] = 0.0;
    __syncthreads();

    const unsigned wave_in_blk = threadIdx.x >> 5;          // 0..7 (wave32)
    const unsigned wave_id = blockIdx.x * (blockDim.x >> 5) + wave_in_blk;
    const uint64_t gaddr = (uint64_t)(uintptr_t)out + (uint64_t)wave_id * CHUNK_BYTES;
    const uint32_t lds_addr = (uint32_t)(uintptr_t)&zbuf[0];  // LDS byte offset

    // ---- D# group 0 (ISA 8.3): count=1, lds_addr, 57-bit global addr, type=2
    v4u g0;
    g0.x = 1u;                                   // count=1, is_restore=0, gather=0
    g0.y = lds_addr;                             // [63:32]  lds_addr
    g0.z = (uint32_t)(gaddr & 0xFFFFFFFFull);    // [95:64]  global_addr lo
    g0.w = (uint32_t)((gaddr >> 32) & 0x01FFFFFFull) | (2u << 30); // addr hi | type=2

    // ---- D# group 1 (ISA 8.4): data_size=8B, 1 x 4096-elem tile ------------
    v8i g1;
    g1.s0 = (int)(3u << 16);                     // workgroup_mask=0, data_size=3 (8B)
    g1.s1 = (int)((CHUNK_DOUBLES & 0xFFFFu) << 16);   // tensor_dim0 lo16 @ [63:48]
    g1.s2 = (int)((CHUNK_DOUBLES >> 16) | (1u << 16)); // tensor_dim0 hi16, tensor_dim1=1
    g1.s3 = (int)((unsigned)CHUNK_DOUBLES << 16);      // tensor_dim1 hi=0, tile_dim0=4096
    g1.s4 = (int)1;                              // tile_dim1=1, tile_dim2=0
    g1.s5 = (int)CHUNK_DOUBLES;                  // tensor_dim0_stride lo32
    g1.s6 = 0;                                   // stride hi / dim1_stride lo
    g1.s7 = 0;

    v4i g2 = {0, 0, 0, 0};                       // <=2D: groups 2/3 unused
    v4i g3 = {0, 0, 0, 0};
    v8i g4 = {0, 0, 0, 0, 0, 0, 0, 0};           // 6-arg form: extra group

    // cpol TH[2:0]=1 -> non-temporal store (256MB stream exceeds 192MB L2).
    __builtin_amdgcn_tensor_store_from_lds(g0, g1, g2, g3, g4, 1);
    __builtin_amdgcn_s_wait_tensorcnt(0);
}

// --- Kernel 3: overwrite the 4096 diagonal entries (runs after the fill) ----
__global__ __launch_bounds__(256) void k_writediag(const double* __restrict__ ws,
                                                   double* __restrict__ out) {
    int i = (int)(blockIdx.x * blockDim.x + threadIdx.x);
    if (i < DIM) {
        double reg = ws[DIM];
        double v = ws[i] + reg;
        size_t o = (size_t)i * (DIM + 1) * 2;    // (i,i) complex -> 2 doubles
        out[o]     = v;
        out[o + 1] = 0.0;
    }
}

extern "C" void kernel_launch(void* const* d_in, const int* in_sizes, int n_in,
                              void* d_out, int out_size, void* d_ws, size_t ws_size,
                              hipStream_t stream) {
    (void)in_sizes; (void)n_in; (void)out_size; (void)ws_size;
    const double* s_real = (const double*)d_in[0];
    const double* s_imag = (const double*)d_in[1];
    double* out = (double*)d_out;   // complex128 -> interleaved (re, im) doubles
    double* ws  = (double*)d_ws;    // ws[0..4095] = diag, ws[4096] = reg

    k_diag<<<1, 1024, 0, stream>>>(s_real, s_imag, ws);
    k_zero_tdm<<<N_CHUNKS / 8, 256, 0, stream>>>(out);   // 1024 blocks, 8 waves each
    k_writediag<<<(DIM + 255) / 256, 256, 0, stream>>>(ws, out);
}